// SPENModel_11321533792445
// MI455X (gfx1250) — compile-verified
//
#include <hip/hip_runtime.h>
#include <math.h>

// ---------------------------------------------------------------------------
// Problem constants (match reference)
// ---------------------------------------------------------------------------
#define BATCH   8192
#define INPUTS  1836
#define LABELS  159
#define LPAD    160
#define F1      150
#define HIDDEN  1024
#define ITERS   20
#define EPS     1e-6f

typedef __attribute__((ext_vector_type(2))) float v2f;
typedef __attribute__((ext_vector_type(8))) float v8f;

#define AS1 __attribute__((address_space(1)))
#define AS3 __attribute__((address_space(3)))

#if __has_builtin(__builtin_amdgcn_global_load_async_to_lds_b32)
#define HAVE_ASYNC 1
#else
#define HAVE_ASYNC 0
#endif

// ---------------------------------------------------------------------------
// Generic tiled GEMM:  Y[M x ldy] = postop( X(M x K, row stride ldx, elem
// stride sx) * W(N x K, row-major)^T + bias )  using V_WMMA_F32_16X16X4_F32.
//
// Block tile 128(M) x 64(N), 256 threads = 8 waves (4x2), each wave owns a
// 32x32 region = 2x2 accumulators -> each LDS fragment feeds 2 WMMAs.
// K staged through double-buffered LDS chunks of KC; chunk k+1 is prefetched
// with GLOBAL_LOAD_ASYNC_TO_LDS_B32 (ASYNCcnt) while chunk k is consumed.
// mode: 0 = none, 1 = relu, 2 = scale[col] * sigmoid(acc)
// ---------------------------------------------------------------------------
#define KC   32
#define BM   128
#define BN   64
#define LDSK (KC + 1)

__device__ __forceinline__
void stage_tiles(float (*Xb)[LDSK], float (*Wb)[LDSK],
                 const float* __restrict__ X, const float* __restrict__ W,
                 int bm, int bn, int k0,
                 int M, int N, int K, int ldx, int sx, int ldw, int tid)
{
    // X tile: BM x KC
    for (int i = tid; i < BM * KC; i += 256) {
        int r = i >> 5, c = i & (KC - 1);
        int gr = bm + r, gk = k0 + c;
        float* lp = &Xb[r][c];
        bool ok = (gr < M) && (gk < K);
#if HAVE_ASYNC
        if (ok) {
            const float* gp = &X[(size_t)gr * ldx + (size_t)gk * sx];
            __builtin_amdgcn_global_load_async_to_lds_b32(
                (AS1 int*)(AS1 void*)gp, (AS3 int*)(AS3 void*)lp, 0, 0);
        } else {
            *lp = 0.0f;                      // ds_store zero-fill (DScnt)
        }
#else
        *lp = ok ? X[(size_t)gr * ldx + (size_t)gk * sx] : 0.0f;
#endif
    }
    // W tile: BN x KC
    for (int i = tid; i < BN * KC; i += 256) {
        int r = i >> 5, c = i & (KC - 1);
        int gn = bn + r, gk = k0 + c;
        float* lp = &Wb[r][c];
        bool ok = (gn < N) && (gk < K);
#if HAVE_ASYNC
        if (ok) {
            const float* gp = &W[(size_t)gn * ldw + gk];
            __builtin_amdgcn_global_load_async_to_lds_b32(
                (AS1 int*)(AS1 void*)gp, (AS3 int*)(AS3 void*)lp, 0, 0);
        } else {
            *lp = 0.0f;
        }
#else
        *lp = ok ? W[(size_t)gn * ldw + gk] : 0.0f;
#endif
    }
}

__device__ __forceinline__ void wait_staged()
{
#if HAVE_ASYNC
#if __has_builtin(__builtin_amdgcn_s_wait_asynccnt)
    __builtin_amdgcn_s_wait_asynccnt(0);
#elif defined(__gfx1250__)
    asm volatile("s_wait_asynccnt 0x0" ::: "memory");
#endif
#endif
    __syncthreads();   // also flushes DScnt for the zero-fill stores
}

__global__ __launch_bounds__(256)
void gemm_wmma_f32(const float* __restrict__ X, const float* __restrict__ W,
                   const float* __restrict__ bias, const float* __restrict__ scale,
                   float* __restrict__ Y,
                   int M, int N, int K, int ldx, int sx, int ldw, int ldy, int mode)
{
    __shared__ float Xs[2][BM][LDSK];
    __shared__ float Ws[2][BN][LDSK];

    const int tid  = threadIdx.x;
    const int bm   = blockIdx.x * BM;
    const int bn   = blockIdx.y * BN;
    const int w    = tid >> 5;            // wave id 0..7
    const int lane = tid & 31;
    const int half = lane >> 4;           // 0 or 1
    const int l16  = lane & 15;
    const int wm   = (w & 3) * 32;        // wave M offset in block tile
    const int wn   = (w >> 2) * 32;       // wave N offset in block tile

    v8f acc[2][2];
    acc[0][0] = (v8f){}; acc[0][1] = (v8f){};
    acc[1][0] = (v8f){}; acc[1][1] = (v8f){};

    const int nch = (K + KC - 1) / KC;

    // prefetch chunk 0
    stage_tiles(Xs[0], Ws[0], X, W, bm, bn, 0, M, N, K, ldx, sx, ldw, tid);

    for (int ci = 0; ci < nch; ++ci) {
        wait_staged();                    // chunk ci resident & visible
        if (ci + 1 < nch)                 // prefetch chunk ci+1 into other buf
            stage_tiles(Xs[(ci + 1) & 1], Ws[(ci + 1) & 1], X, W,
                        bm, bn, (ci + 1) * KC, M, N, K, ldx, sx, ldw, tid);

        float (*Xb)[LDSK] = Xs[ci & 1];
        float (*Wb)[LDSK] = Ws[ci & 1];

        #pragma unroll
        for (int kk = 0; kk < KC; kk += 4) {
            const int kc = kk + 2 * half;
            v2f a0, a1, b0, b1;
            a0.x = Xb[wm + l16][kc];      a0.y = Xb[wm + l16][kc + 1];
            a1.x = Xb[wm + 16 + l16][kc]; a1.y = Xb[wm + 16 + l16][kc + 1];
            b0.x = Wb[wn + l16][kc];      b0.y = Wb[wn + l16][kc + 1];
            b1.x = Wb[wn + 16 + l16][kc]; b1.y = Wb[wn + 16 + l16][kc + 1];
            acc[0][0] = __builtin_amdgcn_wmma_f32_16x16x4_f32(
                            false, a0, false, b0, (short)0, acc[0][0], false, false);
            acc[0][1] = __builtin_amdgcn_wmma_f32_16x16x4_f32(
                            false, a0, false, b1, (short)0, acc[0][1], false, false);
            acc[1][0] = __builtin_amdgcn_wmma_f32_16x16x4_f32(
                            false, a1, false, b0, (short)0, acc[1][0], false, false);
            acc[1][1] = __builtin_amdgcn_wmma_f32_16x16x4_f32(
                            false, a1, false, b1, (short)0, acc[1][1], false, false);
        }
        // next wait_staged()'s barrier separates this compute from the
        // prefetch that will overwrite buf (ci&1) two chunks from now
    }

    // epilogue: C layout -> row = r + 8*half, col = l16 within a 16x16 tile
    #pragma unroll
    for (int nt = 0; nt < 2; ++nt) {
        const int col = bn + wn + nt * 16 + l16;
        if (col >= ldy) continue;
        float bv = (bias  && col < N) ? bias[col]  : 0.0f;
        float sv = (scale && col < N) ? scale[col] : 0.0f;
        #pragma unroll
        for (int mt = 0; mt < 2; ++mt) {
            #pragma unroll
            for (int r = 0; r < 8; ++r) {
                const int row = bm + wm + mt * 16 + r + 8 * half;
                if (row < M) {
                    float v = acc[mt][nt][r] + bv;
                    if (mode == 1)      v = fmaxf(v, 0.0f);
                    else if (mode == 2) v = sv * (1.0f / (1.0f + expf(-v)));
                    Y[(size_t)row * ldy + col] = v;
                }
            }
        }
    }
}

// ---------------------------------------------------------------------------
// Wg1 (HIDDEN x LABELS) -> Wg1T (LPAD x HIDDEN), zero-padded row 159
// ---------------------------------------------------------------------------
__global__ void transpose_wg1(const float* __restrict__ Wg1, float* __restrict__ Wg1T)
{
    int idx = blockIdx.x * blockDim.x + threadIdx.x;
    if (idx >= LPAD * HIDDEN) return;
    int l = idx / HIDDEN, h = idx % HIDDEN;
    Wg1T[idx] = (l < LABELS) ? Wg1[(size_t)h * LABELS + l] : 0.0f;
}

// ---------------------------------------------------------------------------
// Exact JAX threefry-2x32 (key(1) -> ks=(0,1)) for the stick-breaking init.
// uniform(key(1),(159,3)); col0:=0, col2:=1 -> p = (u1, 1-u1) per label.
// ---------------------------------------------------------------------------
__device__ __forceinline__ unsigned rotl32(unsigned x, int r)
{ return (x << r) | (x >> (32 - r)); }

__device__ void threefry2x32(unsigned x0, unsigned x1, unsigned& o0, unsigned& o1)
{
    const unsigned ks0 = 0u, ks1 = 1u;
    const unsigned ks2 = 0x1BD11BDAu ^ ks0 ^ ks1;
    x0 += ks0; x1 += ks1;
    #define TF_R(r) { x0 += x1; x1 = rotl32(x1, (r)); x1 ^= x0; }
    TF_R(13) TF_R(15) TF_R(26) TF_R(6)   x0 += ks1; x1 += ks2 + 1u;
    TF_R(17) TF_R(29) TF_R(16) TF_R(24)  x0 += ks2; x1 += ks0 + 2u;
    TF_R(13) TF_R(15) TF_R(26) TF_R(6)   x0 += ks0; x1 += ks1 + 3u;
    TF_R(17) TF_R(29) TF_R(16) TF_R(24)  x0 += ks1; x1 += ks2 + 4u;
    TF_R(13) TF_R(15) TF_R(26) TF_R(6)   x0 += ks2; x1 += ks0 + 5u;
    #undef TF_R
    o0 = x0; o1 = x1;
}

__global__ void init_pred_kernel(float* __restrict__ pred)
{
    int idx = blockIdx.x * blockDim.x + threadIdx.x;
    if (idx >= BATCH * LABELS) return;
    int l = idx % LABELS;
    // flat index into uniform(key, (159,3)) of element (l, 1); count=477 (odd)
    unsigned i = 3u * (unsigned)l + 1u;
    unsigned c0, c1, y0, y1;
    bool take_first;
    if (i < 239u) { c0 = i;        c1 = (i == 238u) ? 0u : i + 239u; take_first = true; }
    else          { c0 = i - 239u; c1 = i;                           take_first = false; }
    threefry2x32(c0, c1, y0, y1);
    unsigned bits = take_first ? y0 : y1;
    float u = __uint_as_float((bits >> 9) | 0x3f800000u) - 1.0f;  // [0,1)
    pred[(size_t)idx * 2 + 0] = u;         // sorted diffs of (0, u, 1)
    pred[(size_t)idx * 2 + 1] = 1.0f - u;
}

// ---------------------------------------------------------------------------
// One mirror-descent step:
//   g0 = 0            - (log(p0+e) + p0/(p0+e))
//   g1 = logits + gle - (log(p1+e) + p1/(p1+e))
//   lr = 1/sqrt(it); stabilized multiplicative update + renormalize
// ---------------------------------------------------------------------------
__global__ void update_kernel(const float* __restrict__ pred_in,
                              float* __restrict__ pred_out,
                              const float* __restrict__ logits,
                              const float* __restrict__ G1,
                              int it)
{
    int idx = blockIdx.x * blockDim.x + threadIdx.x;
    if (idx >= BATCH * LABELS) return;
    int b = idx / LABELS, l = idx % LABELS;

    float p0 = pred_in[(size_t)idx * 2 + 0];
    float p1 = pred_in[(size_t)idx * 2 + 1];
    float pot1 = logits[(size_t)b * LPAD + l];
    float gle  = G1[(size_t)b * LPAD + l];

    float g0 = -(logf(p0 + EPS) + p0 / (p0 + EPS));
    float g1 = pot1 + gle - (logf(p1 + EPS) + p1 / (p1 + EPS));

    float lr  = rsqrtf((float)it);
    float lg0 = lr * g0, lg1 = lr * g1;
    float m   = fmaxf(lg0, lg1);
    float e0  = p0 * expf(lg0 - m);
    float e1  = p1 * expf(lg1 - m);
    float s   = e0 + e1 + EPS;
    pred_out[(size_t)idx * 2 + 0] = e0 / s;
    pred_out[(size_t)idx * 2 + 1] = e1 / s;
}

// ---------------------------------------------------------------------------
// Host-side orchestration
// ---------------------------------------------------------------------------
extern "C" void kernel_launch(void* const* d_in, const int* in_sizes, int n_in,
                              void* d_out, int out_size, void* d_ws, size_t ws_size,
                              hipStream_t stream)
{
    (void)in_sizes; (void)n_in; (void)out_size; (void)ws_size;

    const float* xs  = (const float*)d_in[0];
    const float* Wf1 = (const float*)d_in[1];
    const float* bf1 = (const float*)d_in[2];
    const float* Wf2 = (const float*)d_in[3];
    const float* bf2 = (const float*)d_in[4];
    const float* Wf3 = (const float*)d_in[5];
    const float* bf3 = (const float*)d_in[6];
    const float* Wg1 = (const float*)d_in[7];
    const float* Wg2 = (const float*)d_in[8];
    float* out = (float*)d_out;

    // workspace layout (floats).  h1/h2 alias the S region (dead before loop).
    float* ws = (float*)d_ws;
    const size_t szS = (size_t)BATCH * HIDDEN;
    float* S      = ws;                                  // sigma(z)*Wg2
    float* h1     = ws;                                  // alias (MLP phase only)
    float* h2     = ws + (size_t)BATCH * LPAD;           // alias (MLP phase only)
    float* logits = ws + szS;
    float* pred   = logits + (size_t)BATCH * LPAD;
    float* G1     = pred + (size_t)BATCH * LABELS * 2;
    float* Wg1T   = G1 + (size_t)BATCH * LPAD;

    const dim3 blk(256);
    const dim3 gM((BATCH + BM - 1) / BM, (LPAD + BN - 1) / BN);    // 64 x 3
    const dim3 gH((BATCH + BM - 1) / BM, (HIDDEN + BN - 1) / BN);  // 64 x 16
    const int  nEl = BATCH * LABELS;
    const dim3 gEl((nEl + 255) / 256);

    // Wg1^T (iteration-invariant)
    transpose_wg1<<<(LPAD * HIDDEN + 255) / 256, blk, 0, stream>>>(Wg1, Wg1T);

    // feature MLP -> potentials channel 1
    gemm_wmma_f32<<<gM, blk, 0, stream>>>(xs, Wf1, bf1, nullptr, h1,
                                          BATCH, F1, INPUTS, INPUTS, 1, INPUTS, LPAD, 1);
    gemm_wmma_f32<<<gM, blk, 0, stream>>>(h1, Wf2, bf2, nullptr, h2,
                                          BATCH, F1, F1, LPAD, 1, F1, LPAD, 1);
    gemm_wmma_f32<<<gM, blk, 0, stream>>>(h2, Wf3, bf3, nullptr, logits,
                                          BATCH, LABELS, F1, LPAD, 1, F1, LPAD, 0);

    // exact stick-breaking init (JAX threefry, key(1))
    init_pred_kernel<<<gEl, blk, 0, stream>>>(pred);

    // 19 mirror-descent iterations (it = 1..19)
    for (int it = 1; it < ITERS; ++it) {
        // S = Wg2 .* sigmoid( pred[:,:,1] @ Wg1^T )   (strided X: sx=2)
        gemm_wmma_f32<<<gH, blk, 0, stream>>>(pred + 1, Wg1, nullptr, Wg2, S,
                                              BATCH, HIDDEN, LABELS,
                                              LABELS * 2, 2, LABELS, HIDDEN, 2);
        // G1 = S @ Wg1  (via Wg1T row-major)
        gemm_wmma_f32<<<gM, blk, 0, stream>>>(S, Wg1T, nullptr, nullptr, G1,
                                              BATCH, LABELS, HIDDEN,
                                              HIDDEN, 1, HIDDEN, LPAD, 0);
        // multiplicative update; last iteration writes straight to d_out
        float* dst = (it == ITERS - 1) ? out : pred;
        update_kernel<<<gEl, blk, 0, stream>>>(pred, dst, logits, G1, it);
    }
}